// Q0DistMultiTask_24275155157424
// MI455X (gfx1250) — compile-verified
//
#include <hip/hip_runtime.h>
#include <hip/hip_bf16.h>
#include <stdint.h>

typedef __attribute__((ext_vector_type(2))) float v2f;
typedef __attribute__((ext_vector_type(8))) float v8f;

#define S_TASKS 32
#define T_TIME 128
#define NSAMP 128
#define NT 64  // num_tasks

// Async global->LDS staging path (CDNA5 GLOBAL_LOAD_ASYNC_TO_LDS_B128, ASYNCcnt).
// Probe round 2: builtin takes (int4vec*, int4vec*, int, int).
#if defined(__gfx1250__) && __has_builtin(__builtin_amdgcn_global_load_async_to_lds_b128) && \
    __has_builtin(__builtin_amdgcn_s_wait_asynccnt)
#define USE_ASYNC_LDS 1
typedef int v4i_vs __attribute__((vector_size(16)));
#else
#define USE_ASYNC_LDS 0
#endif

// ---------------------------------------------------------------------------
// Kernel 1: B_b[i][j] = sum_k L[ids[i],k] * L[ids[j],k]   (32x32, K=64)
// ---------------------------------------------------------------------------
__global__ __launch_bounds__(1024) void bb_kernel(const int* __restrict__ ids,
                                                  const float* __restrict__ L,
                                                  float* __restrict__ Bb) {
  const int i = threadIdx.y;
  const int j = threadIdx.x;
  const float* Li = L + ids[i] * NT;
  const float* Lj = L + ids[j] * NT;
  float acc = 0.0f;
#pragma unroll
  for (int k = 0; k < NT; ++k) acc = fmaf(Li[k], Lj[k], acc);
  Bb[i * S_TASKS + j] = acc;
}

// ---------------------------------------------------------------------------
// Kernel 2: 32x32 Cholesky of B_b (+1e-6 jitter, absorbing the kron jitter)
// ---------------------------------------------------------------------------
__global__ __launch_bounds__(32) void chol32_kernel(const float* __restrict__ Bb,
                                                    float* __restrict__ LA) {
  __shared__ float A[S_TASKS * S_TASKS];
  const int tid = threadIdx.x;
  for (int j = 0; j < S_TASKS; ++j) A[tid * S_TASKS + j] = Bb[tid * S_TASKS + j];
  A[tid * S_TASKS + tid] += 1e-6f;
  __syncthreads();
  for (int k = 0; k < S_TASKS; ++k) {
    if (tid == 0) A[k * S_TASKS + k] = sqrtf(A[k * S_TASKS + k]);
    __syncthreads();
    if (tid > k) A[tid * S_TASKS + k] /= A[k * S_TASKS + k];
    __syncthreads();
    if (tid > k) {
      const float aik = A[tid * S_TASKS + k];
      for (int j = k + 1; j <= tid; ++j)
        A[tid * S_TASKS + j] = fmaf(-aik, A[j * S_TASKS + k], A[tid * S_TASKS + j]);
    }
    __syncthreads();
  }
  for (int j = 0; j < S_TASKS; ++j)
    LA[tid * S_TASKS + j] = (j <= tid) ? A[tid * S_TASKS + j] : 0.0f;
}

// ---------------------------------------------------------------------------
// Kernel 3: 128x128 Cholesky of K_time (row-per-thread, 64 KB LDS)
// ---------------------------------------------------------------------------
__global__ __launch_bounds__(128) void chol128_kernel(const float* __restrict__ Kt,
                                                      float* __restrict__ LB) {
  __shared__ float A[T_TIME * T_TIME];  // 64 KB
  const int tid = threadIdx.x;
  for (int j = 0; j < T_TIME; ++j) A[tid * T_TIME + j] = Kt[tid * T_TIME + j];
  __syncthreads();
  for (int k = 0; k < T_TIME; ++k) {
    if (tid == 0) A[k * T_TIME + k] = sqrtf(A[k * T_TIME + k]);
    __syncthreads();
    if (tid > k) A[tid * T_TIME + k] /= A[k * T_TIME + k];
    __syncthreads();
    if (tid > k) {
      const float aik = A[tid * T_TIME + k];
      for (int j = k + 1; j <= tid; ++j)
        A[tid * T_TIME + j] = fmaf(-aik, A[j * T_TIME + k], A[tid * T_TIME + j]);
    }
    __syncthreads();
  }
  for (int j = 0; j < T_TIME; ++j)
    LB[tid * T_TIME + j] = (j <= tid) ? A[tid * T_TIME + j] : 0.0f;
}

// ---------------------------------------------------------------------------
// Kernel 4: eps ~ N(0,1), Threefry-2x32-20 counter RNG + Box-Muller.
// ---------------------------------------------------------------------------
__device__ __forceinline__ void threefry2x32(uint32_t k0, uint32_t k1,
                                             uint32_t c0, uint32_t c1,
                                             uint32_t& o0, uint32_t& o1) {
  const uint32_t rotA[4] = {13u, 15u, 26u, 6u};
  const uint32_t rotB[4] = {17u, 29u, 16u, 24u};
  const uint32_t ks2 = k0 ^ k1 ^ 0x1BD11BDAu;
  const uint32_t ks[3] = {k0, k1, ks2};
  uint32_t x0 = c0 + k0, x1 = c1 + k1;
#pragma unroll
  for (int i = 0; i < 5; ++i) {
    const uint32_t* rot = (i & 1) ? rotB : rotA;
#pragma unroll
    for (int r = 0; r < 4; ++r) {
      x0 += x1;
      x1 = (x1 << rot[r]) | (x1 >> (32u - rot[r]));
      x1 ^= x0;
    }
    x0 += ks[(i + 1) % 3];
    x1 += ks[(i + 2) % 3] + (uint32_t)(i + 1);
  }
  o0 = x0;
  o1 = x1;
}

__global__ __launch_bounds__(256) void geneps_kernel(float* __restrict__ eps) {
  const uint32_t idx = blockIdx.x * blockDim.x + threadIdx.x;  // 262144 threads
  uint32_t o0, o1;
  threefry2x32(0u, 42u, idx, 0x9E3779B9u, o0, o1);
  float u1 = (float)(o0 >> 8) * (1.0f / 16777216.0f);
  float u2 = (float)(o1 >> 8) * (1.0f / 16777216.0f);
  u1 = fmaxf(u1, 5.96e-8f);
  const float r = sqrtf(-2.0f * logf(u1));
  const float th = 6.2831853071795864f * u2;
  eps[2u * idx + 0u] = r * cosf(th);
  eps[2u * idx + 1u] = r * sinf(th);
}

// ---------------------------------------------------------------------------
// Kernel 5: per-sample  out_n = LA (32x32) @ E_n (32x128) @ LB^T (128x128)
// One workgroup (8 waves) per sample. fp32 WMMA 16x16x4.
// Wave w owns N-columns [16w, 16w+16); does both 16-row M tiles.
//   A  vgpr0 = A[lo][kb+2*hi],      vgpr1 = A[lo][kb+2*hi+1]
//   B  vgpr0 = B[kb+2*hi][lo],      vgpr1 = B[kb+2*hi+1][lo]
//   C  vgpr r = C[r + 8*hi][lo]
// ---------------------------------------------------------------------------
__global__ __launch_bounds__(256) void sample_kernel(const float* __restrict__ eps,
                                                     const float* __restrict__ LA,
                                                     const float* __restrict__ LB,
                                                     float* __restrict__ out) {
  __shared__ float sE[S_TASKS * T_TIME];   // 16 KB
  __shared__ float sF[S_TASKS * T_TIME];   // 16 KB
  __shared__ float sLA[S_TASKS * S_TASKS]; // 4 KB
  const int n = blockIdx.x;
  const int tid = threadIdx.x;

  // Pull this WG's slice of LB toward the caches early (global_prefetch_b8).
  __builtin_prefetch(LB + ((tid * 64) & (T_TIME * T_TIME - 1)), 0, 0);

#if USE_ASYNC_LDS
  // Stage E_n (16 KB) and LA (4 KB) via GLOBAL_LOAD_ASYNC_TO_LDS_B128:
  // bypasses VGPRs, tracked by ASYNCcnt. All offsets 16B-aligned.
  {
    const int base = tid * 4;  // 16 bytes per lane per issue
#pragma unroll
    for (int it = 0; it < 4; ++it) {
      const int idx = base + it * 1024;
      __builtin_amdgcn_global_load_async_to_lds_b128(
          (v4i_vs*)(eps + n * (S_TASKS * T_TIME) + idx), (v4i_vs*)(&sE[idx]), 0, 0);
    }
    __builtin_amdgcn_global_load_async_to_lds_b128(
        (v4i_vs*)(LA + base), (v4i_vs*)(&sLA[base]), 0, 0);
    __builtin_amdgcn_s_wait_asynccnt(0);
  }
#else
  for (int idx = tid; idx < S_TASKS * T_TIME; idx += 256)
    sE[idx] = eps[n * (S_TASKS * T_TIME) + idx];
  for (int idx = tid; idx < S_TASKS * S_TASKS; idx += 256)
    sLA[idx] = LA[idx];
#endif
  __syncthreads();

  const int wave = tid >> 5;
  const int lane = tid & 31;
  const int lo = lane & 15;
  const int hi = lane >> 4;
  const int nb = wave * 16;

  // ---- Step 1: F = LA @ E_n  (M=32, N=128, K=32) ----
#pragma unroll
  for (int mt = 0; mt < 2; ++mt) {
    v8f c = {};
#pragma unroll
    for (int kb = 0; kb < S_TASKS; kb += 4) {
      v2f a, b;
      const int k0 = kb + 2 * hi;
      a.x = sLA[(mt * 16 + lo) * S_TASKS + k0];
      a.y = sLA[(mt * 16 + lo) * S_TASKS + k0 + 1];
      b.x = sE[(k0 + 0) * T_TIME + nb + lo];
      b.y = sE[(k0 + 1) * T_TIME + nb + lo];
      c = __builtin_amdgcn_wmma_f32_16x16x4_f32(false, a, false, b, (short)0, c,
                                                false, false);
    }
#pragma unroll
    for (int r = 0; r < 8; ++r)
      sF[(mt * 16 + r + 8 * hi) * T_TIME + nb + lo] = c[r];
  }
  __syncthreads();

  // ---- Step 2: out_n = F @ LB^T  (M=32, N=128, K=128); B[k][t] = LB[t][k] ----
#pragma unroll
  for (int mt = 0; mt < 2; ++mt) {
    v8f c = {};
#pragma unroll
    for (int kb = 0; kb < T_TIME; kb += 4) {
      v2f a, b;
      const int k0 = kb + 2 * hi;
      a.x = sF[(mt * 16 + lo) * T_TIME + k0];
      a.y = sF[(mt * 16 + lo) * T_TIME + k0 + 1];
      const float2 lb2 = *(const float2*)(LB + (nb + lo) * T_TIME + k0);
      b.x = lb2.x;
      b.y = lb2.y;
      c = __builtin_amdgcn_wmma_f32_16x16x4_f32(false, a, false, b, (short)0, c,
                                                false, false);
    }
#pragma unroll
    for (int r = 0; r < 8; ++r)
      out[n * (S_TASKS * T_TIME) + (mt * 16 + r + 8 * hi) * T_TIME + nb + lo] = c[r];
  }
}

// ---------------------------------------------------------------------------
// Launch: inputs are [ids(i32,32), num_samples(i32,1), L(f32,4096), K_time(f32,16384)]
// Workspace (floats): LA @0 (1024) | LB @1024 (16384) | Bb @17408 (1024) |
//                     eps @32768 (524288)  => ~2.18 MB
// ---------------------------------------------------------------------------
extern "C" void kernel_launch(void* const* d_in, const int* in_sizes, int n_in,
                              void* d_out, int out_size, void* d_ws, size_t ws_size,
                              hipStream_t stream) {
  (void)in_sizes; (void)n_in; (void)out_size; (void)ws_size;
  const int*   ids = (const int*)d_in[0];
  const float* L   = (const float*)d_in[2];
  const float* Kt  = (const float*)d_in[3];
  float* W   = (float*)d_ws;
  float* LAp = W;
  float* LBp = W + 1024;
  float* Bbp = W + 1024 + 16384;
  float* epsp = W + 32768;
  float* out = (float*)d_out;

  bb_kernel<<<1, dim3(32, 32, 1), 0, stream>>>(ids, L, Bbp);
  chol32_kernel<<<1, 32, 0, stream>>>(Bbp, LAp);
  chol128_kernel<<<1, 128, 0, stream>>>(Kt, LBp);
  geneps_kernel<<<(NSAMP * S_TASKS * T_TIME / 2) / 256, 256, 0, stream>>>(epsp);
  sample_kernel<<<NSAMP, 256, 0, stream>>>(epsp, LAp, LBp, out);
}